// CapsLayerTwo_10892037063178
// MI455X (gfx1250) — compile-verified
//
#include <hip/hip_runtime.h>
#include <hip/hip_bf16.h>

typedef __attribute__((ext_vector_type(2))) float v2f;
typedef __attribute__((ext_vector_type(8))) float v8f;

#define NUM_CAPS 32
#define NODES    2048
#define C_IN     16
#define C_OUT    32
#define BATCH    32

// ---------------------------------------------------------------------------
// Kernel A: x_hat[c,b,n,o] = sum_i x[b,n,i] * W[c,n,i,o]
// One wave per (c,n): a 32(B) x 16(I) x 32(O) GEMM done as 2x2 WMMA
// f32 16x16x4 tiles over 4 K-steps. Block = 256 threads = 8 waves = 8 nodes.
// ---------------------------------------------------------------------------
__global__ __launch_bounds__(256) void caps_xhat_wmma(
    const float* __restrict__ x, const float* __restrict__ w,
    float* __restrict__ xhat, int c_base)
{
    __shared__ float xs[8][BATCH][18];   // [node][b][i], row padded to 18 dwords

    const int t     = threadIdx.x;
    const int c_rel = blockIdx.x >> 8;
    const int c     = c_base + c_rel;
    const int n0    = (blockIdx.x & 255) * 8;

    // Cooperative, coalesced load of x[:, n0..n0+7, :] (16 KB) into LDS.
    // float4 index f: b = f>>5, n = (f&31)>>2, comp = f&3 -> each wave streams
    // a contiguous 512B burst per iteration.
    #pragma unroll
    for (int r = 0; r < 4; ++r) {
        int f    = r * 256 + t;
        int b    = f >> 5;
        int rem  = f & 31;
        int nn   = rem >> 2;
        int comp = rem & 3;
        const float4 val =
            ((const float4*)(x + ((size_t)b * NODES + (n0 + nn)) * C_IN))[comp];
        float* dst = &xs[nn][b][comp * 4];
        dst[0] = val.x; dst[1] = val.y; dst[2] = val.z; dst[3] = val.w;
    }
    __syncthreads();

    const int wave = t >> 5;
    const int lane = t & 31;
    const int g    = lane >> 4;   // lane half (selects K group / M+8 rows)
    const int hm   = lane & 15;
    const int n    = n0 + wave;

    const float* __restrict__ wn =
        w + (((size_t)c * NODES + n) * C_IN) * C_OUT;   // W[c,n,:,:], rows of 32

    v8f acc00 = {}, acc01 = {}, acc10 = {}, acc11 = {};

    #pragma unroll
    for (int ks = 0; ks < 4; ++ks) {
        const int k0 = ks * 4 + g * 2;   // A/B K layout: lanes16-31 hold K+2
        v2f a0, a1, b0, b1;
        // A (16x4): row m = b index, per-lane K pair
        a0.x = xs[wave][hm][k0];       a0.y = xs[wave][hm][k0 + 1];
        a1.x = xs[wave][16 + hm][k0];  a1.y = xs[wave][16 + hm][k0 + 1];
        // B (4x16): VGPR r = row K=k0+r, col = hm (+16 for second O tile)
        b0.x = wn[(size_t)k0 * C_OUT + hm];
        b0.y = wn[(size_t)(k0 + 1) * C_OUT + hm];
        b1.x = wn[(size_t)k0 * C_OUT + 16 + hm];
        b1.y = wn[(size_t)(k0 + 1) * C_OUT + 16 + hm];

        acc00 = __builtin_amdgcn_wmma_f32_16x16x4_f32(false, a0, false, b0,
                                                      (short)0, acc00, false, false);
        acc01 = __builtin_amdgcn_wmma_f32_16x16x4_f32(false, a0, false, b1,
                                                      (short)0, acc01, false, false);
        acc10 = __builtin_amdgcn_wmma_f32_16x16x4_f32(false, a1, false, b0,
                                                      (short)0, acc10, false, false);
        acc11 = __builtin_amdgcn_wmma_f32_16x16x4_f32(false, a1, false, b1,
                                                      (short)0, acc11, false, false);
    }

    // D layout: VGPR e -> M=e (lanes 0-15) / M=e+8 (lanes 16-31); col N = hm.
    #pragma unroll
    for (int e = 0; e < 8; ++e) {
        int brow = g * 8 + e;
        size_t base0 = (((size_t)c_rel * BATCH + brow)      * NODES + n) * C_OUT;
        size_t base1 = (((size_t)c_rel * BATCH + 16 + brow) * NODES + n) * C_OUT;
        xhat[base0 + hm]      = acc00[e];
        xhat[base0 + 16 + hm] = acc01[e];
        xhat[base1 + hm]      = acc10[e];
        xhat[base1 + 16 + hm] = acc11[e];
    }
}

// ---------------------------------------------------------------------------
// Kernel B: dynamic routing for one (c,b) pair per workgroup.
// x_hat[c,b] (2048x32 fp32) staged once into LDS (pitch 33 -> conflict-free
// for both o-parallel and n-parallel passes); 3 routing iterations run
// entirely from LDS.
// ---------------------------------------------------------------------------
__global__ __launch_bounds__(256) void caps_route(
    const float* __restrict__ xhat, float* __restrict__ out, int c_base)
{
    extern __shared__ float lds[];
    float* xs   = lds;                    // NODES * 33
    float* bsc  = xs  + NODES * 33;       // NODES   routing logits
    float* ebuf = bsc + NODES;            // NODES   exp(b - max)
    float* part = ebuf + NODES;           // 8 * 32  per-wave partial sums
    float* red  = part + 8 * 32;          // 256     block reduction scratch
    float* vvec = red + 256;              // 32      current output vector

    const int t     = threadIdx.x;
    const int b     = blockIdx.x & 31;
    const int c_rel = blockIdx.x >> 5;
    const int c     = c_base + c_rel;
    const int wave  = t >> 5;
    const int lane  = t & 31;

    const float* __restrict__ src =
        xhat + ((size_t)(c_rel * BATCH + b)) * NODES * C_OUT;

    // Stage x_hat[c,b] into LDS (coalesced float4 stream, 256 KB).
    for (int j = 0; j < 64; ++j) {
        int f    = t + 256 * j;         // float4 index, 16384 total
        int n    = f >> 3;
        int comp = f & 7;
        float4 val = ((const float4*)src)[f];
        float* dst = &xs[n * 33 + comp * 4];
        dst[0] = val.x; dst[1] = val.y; dst[2] = val.z; dst[3] = val.w;
    }
    #pragma unroll
    for (int j = 0; j < 8; ++j) bsc[t + 256 * j] = 0.0f;
    __syncthreads();

    for (int it = 0; it < 3; ++it) {
        // ---- softmax max over N ----
        float lm = -3.4e38f;
        #pragma unroll
        for (int j = 0; j < 8; ++j) lm = fmaxf(lm, bsc[t + 256 * j]);
        red[t] = lm; __syncthreads();
        for (int s = 128; s > 0; s >>= 1) {
            if (t < s) red[t] = fmaxf(red[t], red[t + s]);
            __syncthreads();
        }
        float m = red[0]; __syncthreads();

        // ---- exp and sum ----
        float ls = 0.0f;
        #pragma unroll
        for (int j = 0; j < 8; ++j) {
            int n   = t + 256 * j;
            float e = __expf(bsc[n] - m);
            ebuf[n] = e;
            ls += e;
        }
        red[t] = ls; __syncthreads();
        for (int s = 128; s > 0; s >>= 1) {
            if (t < s) red[t] += red[t + s];
            __syncthreads();
        }
        float denom = red[0]; __syncthreads();

        // ---- s[o] = sum_n e_n * xs[n][o]  (wave w covers n in [w*256, w*256+256)) ----
        float acc = 0.0f;
        const int nb = wave * 256;
        for (int n = nb; n < nb + 256; ++n)
            acc += ebuf[n] * xs[n * 33 + lane];
        part[wave * 32 + lane] = acc;
        __syncthreads();

        // ---- wave 0: finish reduction, squash ----
        if (t < 32) {
            float s = 0.0f;
            #pragma unroll
            for (int w2 = 0; w2 < 8; ++w2) s += part[w2 * 32 + t];
            s /= denom;
            float sq = s * s;
            #pragma unroll
            for (int off = 16; off > 0; off >>= 1)
                sq += __shfl_xor(sq, off, 32);      // wave32 butterfly
            float factor = (sq / (1.0f + sq)) / sqrtf(sq + 1e-8f);
            vvec[t] = s * factor;
        }
        __syncthreads();

        // ---- agreement update: b_n += x_hat[n,:] . v ----
        if (it < 2) {
            #pragma unroll
            for (int j = 0; j < 8; ++j) {
                int n = t + 256 * j;
                float dot = 0.0f;
                #pragma unroll
                for (int o = 0; o < 32; ++o) dot += xs[n * 33 + o] * vvec[o];
                bsc[n] += dot;
            }
            __syncthreads();
        }
    }

    if (t < 32) out[((size_t)c * BATCH + b) * C_OUT + t] = vvec[t];
}

// ---------------------------------------------------------------------------
// Host launcher: chunk over capsules so any ws_size >= 8 MiB works.
// All launches on `stream` (graph-capture safe, deterministic).
// ---------------------------------------------------------------------------
extern "C" void kernel_launch(void* const* d_in, const int* in_sizes, int n_in,
                              void* d_out, int out_size, void* d_ws, size_t ws_size,
                              hipStream_t stream)
{
    const float* x = (const float*)d_in[0];   // [B, N, I]
    const float* w = (const float*)d_in[1];   // [C, N, I, O]
    float* out = (float*)d_out;               // [C, B, 1, 1, O]
    float* ws  = (float*)d_ws;                // x_hat chunks [cc, B, N, O]

    const size_t bytes_per_c = (size_t)BATCH * NODES * C_OUT * sizeof(float); // 8 MiB
    int cc = (int)(ws_size / bytes_per_c);
    if (cc < 1) cc = 1;
    if (cc > NUM_CAPS) cc = NUM_CAPS;

    const size_t lds_bytes =
        (size_t)(NODES * 33 + NODES + NODES + 8 * 32 + 256 + 32) * sizeof(float); // ~289 KB

    for (int c0 = 0; c0 < NUM_CAPS; c0 += cc) {
        int nc = (NUM_CAPS - c0 < cc) ? (NUM_CAPS - c0) : cc;
        caps_xhat_wmma<<<dim3(nc * 256), dim3(256), 0, stream>>>(x, w, ws, c0);
        caps_route  <<<dim3(nc * 32),  dim3(256), lds_bytes, stream>>>(ws, out, c0);
    }
}